// Quantizer_54932631716174
// MI455X (gfx1250) — compile-verified
//
#include <hip/hip_runtime.h>
#include <math.h>

typedef __attribute__((ext_vector_type(2))) float    v2f;
typedef __attribute__((ext_vector_type(8))) float    v8f;
typedef __attribute__((ext_vector_type(4))) unsigned v4u;
typedef __attribute__((ext_vector_type(8))) int      v8i_;
typedef __attribute__((ext_vector_type(4))) int      v4i_;

#define NUM_CODES  2048
#define CODE_DEPTH 64
#define D_MODEL    1024
#define NBATCH     8
#define SEQ        2048
#define NTOK       (NBATCH * SEQ)      // 16384
#define DECAY      0.99f

// ---- workspace layout (32-bit element offsets into d_ws) ----
#define WS_Z       0                                    // z normalized: NTOK x 64
#define WS_CN2     (NTOK * CODE_DEPTH)                  // ||cb[k]||^2 : 2048
#define WS_CD      (WS_CN2 + NUM_CODES)                 // code_dist  : 8 x 2048 (+inf init)
#define WS_BIN     (WS_CD + NBATCH * NUM_CODES)         // bincounts  : 2048 (uint)
#define WS_SUMS    (WS_BIN + NUM_CODES)                 // sums       : 2048 x 64
#define WS_COMMIT  (WS_SUMS + NUM_CODES * CODE_DEPTH)   // scalar accumulator
#define WS_IDX     (WS_COMMIT + 1)                      // int indices: NTOK

// ---- output layout (float element offsets into d_out) ----
#define O_HARD     0
#define O_IDX      (NTOK * CODE_DEPTH)
#define O_COMMIT   (O_IDX + NTOK)
#define O_EMBED    (O_COMMIT + 1)
#define O_CB       (O_EMBED + 1)
#define O_CS       (O_CB + NUM_CODES * CODE_DEPTH)
#define O_CSUM     (O_CS + NUM_CODES)

// ============================================================
// Tensor Data Mover: 2-D tile load Global -> LDS.
// D# packing per CDNA5 ISA 8.3/8.4 (group0: count/lds/global/type,
// group1: data_size + pad + dims/strides). Groups 2/3 zero (<=2D).
// pad_interval_code: 0=2,1=4,...,7=256 dwords; pad_amount_code: 0=1..127=128.
// This toolchain (clang-23) exposes the 6-arg builtin:
//   (v4u g0, v8i g1, v4i g2, v4i g3, v8i ext, i32 cpol)
// ============================================================
static __device__ __forceinline__ void
tdm_load_2d(unsigned lds_byte_addr, const void* gptr,
            unsigned tile_w, unsigned tile_h,
            unsigned tensor_w, unsigned tensor_h, unsigned row_stride,
            unsigned pad_en, unsigned pad_interval_code, unsigned pad_amount_code) {
  const unsigned long long ga = (unsigned long long)(uintptr_t)gptr;
  v4u g0;
  g0.x = 1u;                                              // count=1, user descriptor
  g0.y = lds_byte_addr;                                   // lds_addr
  g0.z = (unsigned)ga;                                    // global_addr[31:0]
  g0.w = (unsigned)((ga >> 32) & 0x01FFFFFFu) | (2u << 30); // addr[56:32] | type=2
  v8i_ g1;
  g1[0] = (int)((2u << 16) | (pad_en << 20) |
                (pad_interval_code << 22) | (pad_amount_code << 25)); // data_size=4B
  g1[1] = (int)((tensor_w & 0xFFFFu) << 16);              // tensor_dim0[15:0]
  g1[2] = (int)(((tensor_w >> 16) & 0xFFFFu) | ((tensor_h & 0xFFFFu) << 16));
  g1[3] = (int)(((tensor_h >> 16) & 0xFFFFu) | ((tile_w & 0xFFFFu) << 16));
  g1[4] = (int)(tile_h & 0xFFFFu);                        // tile_dim1 (tile_dim2=0)
  g1[5] = (int)row_stride;                                // tensor_dim0_stride[31:0]
  g1[6] = 0;
  g1[7] = 0;
  v4i_ g2 = {0, 0, 0, 0};
  v4i_ g3 = {0, 0, 0, 0};
  v8i_ gx = {0, 0, 0, 0, 0, 0, 0, 0};
  __builtin_amdgcn_tensor_load_to_lds(g0, g1, g2, g3, gx, 0);
}

// ============================================================
// Kernel 0: init accumulators + per-code squared norms
// ============================================================
__global__ void __launch_bounds__(256)
k_init(const float* __restrict__ cb, float* __restrict__ ws) {
  const int i = blockIdx.x * blockDim.x + threadIdx.x;   // 16384 threads
  if (i < NUM_CODES) {
    const float* r = cb + (size_t)i * CODE_DEPTH;
    float s = 0.f;
    #pragma unroll 8
    for (int d = 0; d < CODE_DEPTH; ++d) s += r[d] * r[d];
    ws[WS_CN2 + i] = s;
    ((unsigned*)ws)[WS_BIN + i] = 0u;
  }
  if (i < NBATCH * NUM_CODES) ws[WS_CD + i] = __uint_as_float(0x7F800000u);
  for (int j = i; j < NUM_CODES * CODE_DEPTH; j += gridDim.x * blockDim.x)
    ws[WS_SUMS + j] = 0.f;
  if (i == 0) ws[WS_COMMIT] = 0.f;
}

// ============================================================
// Kernel 1: z = normalize_rows(x @ W^T + b) via WMMA f32 16x16x4.
// 128 threads (4 waves), 16 tokens/block; wave w owns depth tile 16w.
// x staged through LDS with double-buffered TDM (16x256 chunks,
// row-padded to 258 dwords => conflict-free ds_load_b64 A-frags).
// ============================================================
#define XCHUNK  256
#define XSTRIDE 258
__global__ void __launch_bounds__(128)
k_down(const float* __restrict__ x, const float* __restrict__ W,
       const float* __restrict__ bias, float* __restrict__ ws) {
  __shared__ float xbuf[2][16 * XSTRIDE + 8];
  __shared__ float zt[16][65];
  __shared__ float rn[16];

  const int t0   = blockIdx.x * 16;
  const int wave = threadIdx.x >> 5;
  const int lane = threadIdx.x & 31;
  const int half = lane >> 4;        // K sub-block selector (ISA A/B layout)
  const int l    = lane & 15;        // row (A) / col (B) within tile
  const int d0   = wave * 16;

  // B fragment cols: depth d0+l; B[k][n] = W[n][k] (W row-major [64][1024])
  const float* wb = W + (size_t)(d0 + l) * D_MODEL + half * 2;

  if (wave == 0)
    tdm_load_2d((unsigned)(uintptr_t)&xbuf[0][0], x + (size_t)t0 * D_MODEL,
                XCHUNK, 16, D_MODEL, 16, D_MODEL, 1u, 7u, 1u);

  v8f acc = {};
  for (int c = 0; c < 4; ++c) {
    if (wave == 0) {
      if (c < 3) {
        tdm_load_2d((unsigned)(uintptr_t)&xbuf[(c + 1) & 1][0],
                    x + (size_t)t0 * D_MODEL + (c + 1) * XCHUNK,
                    XCHUNK, 16, D_MODEL, 16, D_MODEL, 1u, 7u, 1u);
        __builtin_amdgcn_s_wait_tensorcnt(1);   // oldest (chunk c) complete
      } else {
        __builtin_amdgcn_s_wait_tensorcnt(0);
      }
    }
    __syncthreads();

    const float* xb = &xbuf[c & 1][l * XSTRIDE + half * 2];
    const float* wc = wb + c * XCHUNK;
    for (int k = 0; k < XCHUNK; k += 4) {
      v2f a = *(const v2f*)(xb + k);            // ds_load_b64, conflict-free
      v2f bb; bb.x = wc[k]; bb.y = wc[k + 1];
      acc = __builtin_amdgcn_wmma_f32_16x16x4_f32(false, a, false, bb,
                                                  (short)0, acc, false, false);
    }
    __syncthreads();   // all waves done with xbuf[c&1] before it is reloaded
  }

  // C layout: lanes 0-15 rows M=v, lanes 16-31 rows M=v+8; N = l
  const float bd = bias[d0 + l];
  #pragma unroll
  for (int v = 0; v < 8; ++v)
    zt[v + half * 8][d0 + l] = acc[v] + bd;
  __syncthreads();

  if (threadIdx.x < 16) {
    float s = 0.f;
    #pragma unroll 8
    for (int d = 0; d < CODE_DEPTH; ++d) { float v = zt[threadIdx.x][d]; s += v * v; }
    rn[threadIdx.x] = rsqrtf(s);
  }
  __syncthreads();

  for (int i = threadIdx.x; i < 16 * CODE_DEPTH; i += 128) {
    int r = i >> 6, c = i & 63;
    ws[WS_Z + (size_t)(t0 + r) * CODE_DEPTH + c] = zt[r][c] * rn[r];
  }
}

// ============================================================
// Kernel 2: distances + argmin + per-code min + bincounts.
// 256 threads (8 waves), 128 tokens/block. z tile staged by one TDM
// load with 4-dword row padding (stride 68 => conflict-free, 16B rows).
// d2 = 1 + ||c||^2 - 2 z.c ; dist = sqrt(max(d2,0))
// ============================================================
#define ZSTRIDE 68
__global__ void __launch_bounds__(256)
k_assign(const float* __restrict__ cb, float* __restrict__ ws,
         float* __restrict__ out) {
  __shared__ float zt[128 * ZSTRIDE + 8];

  const int t0 = blockIdx.x * 128;
  const int b  = t0 / SEQ;           // uniform per block (128 | 2048)

  const int wave = threadIdx.x >> 5;
  const int lane = threadIdx.x & 31;
  const int half = lane >> 4;
  const int l    = lane & 15;
  const int ts   = wave * 16;

  if (wave == 0) {
    tdm_load_2d((unsigned)(uintptr_t)&zt[0], ws + WS_Z + (size_t)t0 * CODE_DEPTH,
                CODE_DEPTH, 128, CODE_DEPTH, 128, CODE_DEPTH, 1u, 5u, 3u);
    __builtin_amdgcn_s_wait_tensorcnt(0);
  }
  __syncthreads();

  const float* cn2 = ws + WS_CN2;
  unsigned* cdst = (unsigned*)ws + WS_CD + b * NUM_CODES;

  float minv[8]; int mini[8];
  #pragma unroll
  for (int v = 0; v < 8; ++v) { minv[v] = 3.0e38f; mini[v] = 0; }

  const float* arow = &zt[(ts + l) * ZSTRIDE + half * 2];

  for (int c0 = 0; c0 < 2047; c0 += 16) {
    const int  n0    = c0 + l;              // 0-based code (codes = cb[1:])
    const bool valid = (n0 < 2047);
    const int  n     = valid ? n0 : 2046;   // clamp to avoid OOB load
    const float* crow = cb + (size_t)(n + 1) * CODE_DEPTH + half * 2;

    v8f acc = {};
    #pragma unroll
    for (int k = 0; k < CODE_DEPTH; k += 4) {
      v2f a = *(const v2f*)(arow + k);      // ds_load_b64
      v2f bb; bb.x = crow[k]; bb.y = crow[k + 1];
      acc = __builtin_amdgcn_wmma_f32_16x16x4_f32(false, a, false, bb,
                                                  (short)0, acc, false, false);
    }

    const float c2 = cn2[n + 1];
    float mrow = 3.0e38f;
    #pragma unroll
    for (int v = 0; v < 8; ++v) {
      float d2   = 1.0f + c2 - 2.0f * acc[v];
      float dist = sqrtf(fmaxf(d2, 0.0f));
      if (valid && dist < minv[v]) { minv[v] = dist; mini[v] = n0; }
      mrow = fminf(mrow, dist);
    }
    // combine the two 8-token halves (same code n0 in both), one atomic/code
    mrow = fminf(mrow, __shfl_xor(mrow, 16, 32));
    if (valid && half == 0)
      atomicMin(&cdst[n0 + 1], __float_as_uint(mrow));   // dist>=0 => bit order ok
  }

  // per-token argmin across the 16 lanes of each half (C-matrix N dim)
  #pragma unroll
  for (int v = 0; v < 8; ++v) {
    float mv = minv[v]; int mi = mini[v];
    #pragma unroll
    for (int off = 8; off > 0; off >>= 1) {
      float ov = __shfl_xor(mv, off, 16);
      int   oi = __shfl_xor(mi, off, 16);
      if (ov < mv || (ov == mv && oi < mi)) { mv = ov; mi = oi; }
    }
    if (l == 0) {
      const int tok = t0 + ts + v + half * 8;
      const int ci  = mi + 1;                 // reference adds 1
      ((int*)ws)[WS_IDX + tok] = ci;
      out[O_IDX + tok] = (float)ci;
      atomicAdd((unsigned*)ws + WS_BIN + ci, 1u);
    }
  }
}

// ============================================================
// Kernel 3: hard = cb[idx]; straight-through output; commitment
// loss accumulation; scatter sums[idx] += z. One wave per token.
// ============================================================
__global__ void __launch_bounds__(256)
k_gather(const float* __restrict__ cb, float* __restrict__ ws,
         float* __restrict__ out) {
  const int wave = threadIdx.x >> 5;
  const int lane = threadIdx.x & 31;
  const int tok  = blockIdx.x * 8 + wave;
  const int ci   = ((const int*)ws)[WS_IDX + tok];
  const int j    = lane * 2;

  const float h0 = cb[(size_t)ci * CODE_DEPTH + j];
  const float h1 = cb[(size_t)ci * CODE_DEPTH + j + 1];
  const float z0 = ws[WS_Z + (size_t)tok * CODE_DEPTH + j];
  const float z1 = ws[WS_Z + (size_t)tok * CODE_DEPTH + j + 1];

  out[O_HARD + (size_t)tok * CODE_DEPTH + j]     = h0;  // z + sg(hard - z) == hard
  out[O_HARD + (size_t)tok * CODE_DEPTH + j + 1] = h1;

  float d0 = h0 - z0, d1 = h1 - z1;
  float s = d0 * d0 + d1 * d1;
  #pragma unroll
  for (int off = 16; off > 0; off >>= 1) s += __shfl_xor(s, off, 32);
  if (lane == 0) atomicAdd(ws + WS_COMMIT, s);

  atomicAdd(ws + WS_SUMS + (size_t)ci * CODE_DEPTH + j,     z0);
  atomicAdd(ws + WS_SUMS + (size_t)ci * CODE_DEPTH + j + 1, z1);
}

// ============================================================
// Kernel 4: EMA updates, centroids, codebook replace, losses.
// Single 1024-thread block; each thread owns codes {tid, tid+1024}.
// ============================================================
__global__ void __launch_bounds__(1024)
k_final(const float* __restrict__ cb, const float* __restrict__ cs_in,
        const float* __restrict__ csum_in, float* __restrict__ ws,
        float* __restrict__ out) {
  __shared__ float red[1024];
  __shared__ float bc_n, bc_cnt, bc_emb;
  const int tid = threadIdx.x;
  const unsigned* bins = (const unsigned*)ws + WS_BIN;

  float nPart = 0.f, cntPart = 0.f, embPart = 0.f;
  float ncs[2];
  #pragma unroll
  for (int p = 0; p < 2; ++p) {
    const int k = tid + p * 1024;
    const unsigned bc = bins[k];
    const float v = DECAY * cs_in[k] + (1.f - DECAY) * (float)bc;
    ncs[p] = v;
    out[O_CS + k] = v;
    nPart += v;
    if (k >= 1 && bc == 0u) {
      cntPart += 1.f;
      #pragma unroll
      for (int bb = 0; bb < NBATCH; ++bb)
        embPart += ws[WS_CD + bb * NUM_CODES + k];
    }
  }

  red[tid] = nPart; __syncthreads();
  for (int s = 512; s > 0; s >>= 1) { if (tid < s) red[tid] += red[tid + s]; __syncthreads(); }
  if (tid == 0) bc_n = red[0];
  __syncthreads();
  red[tid] = cntPart; __syncthreads();
  for (int s = 512; s > 0; s >>= 1) { if (tid < s) red[tid] += red[tid + s]; __syncthreads(); }
  if (tid == 0) bc_cnt = red[0];
  __syncthreads();
  red[tid] = embPart; __syncthreads();
  for (int s = 512; s > 0; s >>= 1) { if (tid < s) red[tid] += red[tid + s]; __syncthreads(); }
  if (tid == 0) bc_emb = red[0];
  __syncthreads();

  if (tid == 0) {
    const float cnt = bc_cnt * (float)NBATCH;
    out[O_EMBED]  = bc_emb / fmaxf(cnt, 1.0f);
    out[O_COMMIT] = ws[WS_COMMIT] / (float)(NTOK * CODE_DEPTH);
  }

  const float n      = bc_n;
  const float sscale = n / (n + (float)NUM_CODES * 1e-5f);

  #pragma unroll
  for (int p = 0; p < 2; ++p) {
    const int k = tid + p * 1024;
    const unsigned bc = bins[k];
    const float smoothed = (ncs[p] + 1e-5f) * sscale;
    const bool used = (bc != 0u);
    for (int d = 0; d < CODE_DEPTH; ++d) {
      const size_t o = (size_t)k * CODE_DEPTH + d;
      const float s = DECAY * csum_in[o] + (1.f - DECAY) * ws[WS_SUMS + o];
      out[O_CSUM + o] = s;
      float nb = used ? (s / smoothed) : cb[o];
      if (k == 0) nb = 0.f;
      out[O_CB + o] = nb;
    }
  }
}

// ============================================================
extern "C" void kernel_launch(void* const* d_in, const int* in_sizes, int n_in,
                              void* d_out, int out_size, void* d_ws, size_t ws_size,
                              hipStream_t stream) {
  const float* x    = (const float*)d_in[0];   // (8,2048,1024)
  const float* W    = (const float*)d_in[1];   // (64,1024)
  const float* bias = (const float*)d_in[2];   // (64,)
  const float* cb   = (const float*)d_in[3];   // (1,2048,64)
  const float* cs   = (const float*)d_in[4];   // (1,2048)
  const float* csum = (const float*)d_in[5];   // (1,2048,64)
  float* out = (float*)d_out;
  float* ws  = (float*)d_ws;

  k_init  <<<64,         256, 0, stream>>>(cb, ws);
  k_down  <<<NTOK / 16,  128, 0, stream>>>(x, W, bias, ws);
  k_assign<<<NTOK / 128, 256, 0, stream>>>(cb, ws, out);
  k_gather<<<NTOK / 8,   256, 0, stream>>>(cb, ws, out);
  k_final <<<1,         1024, 0, stream>>>(cb, cs, csum, ws, out);
}